// IouLoss_37409165148579
// MI455X (gfx1250) — compile-verified
//
#include <hip/hip_runtime.h>
#include <hip/hip_bf16.h>
#include <math.h>

#define BB 8
#define PP 2048
#define CC 21
#define CENTER_VAR 0.1f
#define SIZE_VAR   0.2f
#define EPSV       1e-8f

typedef __attribute__((ext_vector_type(2))) float v2f;
typedef __attribute__((ext_vector_type(8))) float v8f;

// ---------------------------------------------------------------------------
// Kernel 1: softmax stats -> masks, decode boxes, precompute target areas.
// One thread per (b,p). 16384 threads total.
// ---------------------------------------------------------------------------
__global__ void __launch_bounds__(256)
prep_kernel(const float* __restrict__ conf,
            const float* __restrict__ loc,
            const int*   __restrict__ tconf,
            const float* __restrict__ tloc,
            const float* __restrict__ priors,
            float4* __restrict__ decA,
            float4* __restrict__ decT,
            float*  __restrict__ maskI,
            float*  __restrict__ maskT,
            float*  __restrict__ areaT)
{
    int idx = blockIdx.x * blockDim.x + threadIdx.x;
    if (idx >= BB * PP) return;
    int p = idx & (PP - 1);

    // softmax max-prob and argmax (first max, matching jnp.argmax)
    const float* c = conf + (size_t)idx * CC;
    float m = c[0];
    int am = 0;
#pragma unroll
    for (int i = 1; i < CC; ++i) {
        float v = c[i];
        if (v > m) { m = v; am = i; }
    }
    float se = 0.f;
#pragma unroll
    for (int i = 0; i < CC; ++i) se += expf(c[i] - m);
    float maxp = 1.0f / se;               // exp(m-m)/sum = 1/sum

    bool pt = tconf[idx] > 0;
    bool pi = pt && (maxp > 0.5f) && (am > 0);
    maskT[idx] = pt ? 1.f : 0.f;
    maskI[idx] = pi ? 1.f : 0.f;

    float4 pr = ((const float4*)priors)[p];

    // decode predictions
    {
        float4 l = ((const float4*)loc)[idx];
        float cx = pr.x + l.x * CENTER_VAR * pr.z;
        float cy = pr.y + l.y * CENTER_VAR * pr.w;
        float w  = pr.z * expf(l.z * SIZE_VAR);
        float h  = pr.w * expf(l.w * SIZE_VAR);
        float x0 = cx - 0.5f * w, y0 = cy - 0.5f * h;
        decA[idx] = make_float4(x0, y0, x0 + w, y0 + h);
    }
    // decode targets
    {
        float4 l = ((const float4*)tloc)[idx];
        float cx = pr.x + l.x * CENTER_VAR * pr.z;
        float cy = pr.y + l.y * CENTER_VAR * pr.w;
        float w  = pr.z * expf(l.z * SIZE_VAR);
        float h  = pr.w * expf(l.w * SIZE_VAR);
        float x0 = cx - 0.5f * w, y0 = cy - 0.5f * h;
        float4 t = make_float4(x0, y0, x0 + w, y0 + h);
        decT[idx]  = t;
        areaT[idx] = (t.z - t.x) * (t.w - t.y);
    }
}

// ---------------------------------------------------------------------------
// Kernel 2: per batch, tile of 256 predictions vs all 2048 targets (in LDS).
// grid = (PP/256, BB). Wave partial sums reduced with V_WMMA_F32_16X16X4_F32.
// ---------------------------------------------------------------------------
__global__ void __launch_bounds__(256)
pair_kernel(const float4* __restrict__ decA,
            const float4* __restrict__ decT,
            const float*  __restrict__ maskI,
            const float*  __restrict__ maskT,
            const float*  __restrict__ areaT,
            float* __restrict__ blockSums)
{
    __shared__ float4 sT[PP];     // 32 KB
    __shared__ float  sA[PP];     //  8 KB
    __shared__ float  sM[PP];     //  8 KB
    __shared__ float  waveSums[16];

    const int tid  = threadIdx.x;
    const int b    = blockIdx.y;
    const int base = b * PP;

    for (int i = tid; i < PP; i += 256) {
        sT[i] = decT[base + i];
        sA[i] = areaT[base + i];
        sM[i] = maskT[base + i];
    }
    __syncthreads();

    const int a  = base + blockIdx.x * 256 + tid;
    const float4 A  = decA[a];
    const float  mI = maskI[a];
    const float  areaA = (A.z - A.x) * (A.w - A.y);

    float acc = 0.f;
#pragma unroll 4
    for (int t = 0; t < PP; ++t) {
        const float4 T = sT[t];
        float iw = fmaxf(fminf(A.z, T.z) - fmaxf(A.x, T.x), 0.f);
        float ih = fmaxf(fminf(A.w, T.w) - fmaxf(A.y, T.y), 0.f);
        float inter = iw * ih;
        float uni   = areaA + sA[t] - inter;
        float iou   = inter / fmaxf(uni, EPSV);
        float ew = fmaxf(A.z, T.z) - fminf(A.x, T.x);
        float eh = fmaxf(A.w, T.w) - fminf(A.y, T.y);
        float enc = ew * eh;
        float g = iou - (enc - uni) / fmaxf(enc, EPSV);
        acc = fmaf(sM[t], g, acc);
    }
    acc *= mI;

    // Wave-level reduction via WMMA: A-frag holds per-lane partials, B = ones,
    // C = 0. D row-sums: lane 0 covers M=0..7, lane 16 covers M=8..15.
    v2f afrag; afrag.x = acc;  afrag.y = 0.f;
    v2f bfrag; bfrag.x = 1.f;  bfrag.y = 1.f;
    v8f cfrag = {};
    v8f d = __builtin_amdgcn_wmma_f32_16x16x4_f32(
        false, afrag, false, bfrag, (short)0, cfrag, false, false);
    float x = d[0] + d[1] + d[2] + d[3] + d[4] + d[5] + d[6] + d[7];

    const int lane = tid & 31;
    const int wave = tid >> 5;
    if (lane == 0)  waveSums[wave * 2 + 0] = x;
    if (lane == 16) waveSums[wave * 2 + 1] = x;
    __syncthreads();

    if (tid == 0) {
        float s = 0.f;
#pragma unroll
        for (int i = 0; i < 16; ++i) s += waveSums[i];
        blockSums[b * 8 + blockIdx.x] = s;
    }
}

// ---------------------------------------------------------------------------
// Kernel 3: counts + final scalar. Single block; deterministic.
// ---------------------------------------------------------------------------
__global__ void __launch_bounds__(256)
finalize_kernel(const float* __restrict__ maskI,
                const float* __restrict__ maskT,
                const float* __restrict__ blockSums,
                float* __restrict__ out)
{
    __shared__ int cntI[BB];
    __shared__ int cntT[BB];
    const int tid = threadIdx.x;
    if (tid < BB) { cntI[tid] = 0; cntT[tid] = 0; }
    __syncthreads();

    for (int i = tid; i < BB * PP; i += 256) {
        int b = i >> 11;                       // / PP
        if (maskI[i] > 0.5f) atomicAdd(&cntI[b], 1);
        if (maskT[i] > 0.5f) atomicAdd(&cntT[b], 1);
    }
    __syncthreads();

    if (tid == 0) {
        float total = 0.f, num_boxes = 0.f;
        for (int b = 0; b < BB; ++b) {
            float s = 0.f;
            for (int k = 0; k < 8; ++k) s += blockSums[b * 8 + k];
            float nt = (float)cntT[b];
            float na = (float)cntI[b];
            bool bt = nt > 0.f, ba = na > 0.f;
            float term;
            if (bt && ba)      term = nt - s / fmaxf(nt, 1.f);
            else if (bt != ba) term = 1.f;
            else               term = 0.f;
            total     += term;
            num_boxes += nt;
        }
        out[0] = total / fmaxf(num_boxes, 1.f);
    }
}

// ---------------------------------------------------------------------------
extern "C" void kernel_launch(void* const* d_in, const int* in_sizes, int n_in,
                              void* d_out, int out_size, void* d_ws, size_t ws_size,
                              hipStream_t stream)
{
    const float* conf   = (const float*)d_in[0];   // [B,P,21]
    const float* loc    = (const float*)d_in[1];   // [B,P,4]
    const int*   tconf  = (const int*)  d_in[2];   // [B,P]
    const float* tloc   = (const float*)d_in[3];   // [B,P,4]
    const float* priors = (const float*)d_in[4];   // [P,4]
    float* out = (float*)d_out;

    char* ws = (char*)d_ws;
    float4* decA  = (float4*)ws;                 ws += (size_t)BB * PP * sizeof(float4);
    float4* decT  = (float4*)ws;                 ws += (size_t)BB * PP * sizeof(float4);
    float*  maskI = (float*)ws;                  ws += (size_t)BB * PP * sizeof(float);
    float*  maskT = (float*)ws;                  ws += (size_t)BB * PP * sizeof(float);
    float*  areaT = (float*)ws;                  ws += (size_t)BB * PP * sizeof(float);
    float*  blockSums = (float*)ws;              // 64 floats

    prep_kernel<<<(BB * PP + 255) / 256, 256, 0, stream>>>(
        conf, loc, tconf, tloc, priors, decA, decT, maskI, maskT, areaT);

    pair_kernel<<<dim3(PP / 256, BB), 256, 0, stream>>>(
        decA, decT, maskI, maskT, areaT, blockSums);

    finalize_kernel<<<1, 256, 0, stream>>>(maskI, maskT, blockSums, out);
}